// TGDModel_58617713656424
// MI455X (gfx1250) — compile-verified
//
#include <hip/hip_runtime.h>
#include <hip/hip_bf16.h>

#define NNODES 20000
#define NEDGES 640000

typedef __attribute__((ext_vector_type(2))) float v2f;
typedef __attribute__((ext_vector_type(8))) float v8f;

struct TxPtrs { const float* p[4]; };

static __device__ __forceinline__ int gtid() {
    return blockIdx.x * blockDim.x + threadIdx.x;
}

__global__ void k_zero(float* __restrict__ p, int n) {
    int i = gtid(); if (i < n) p[i] = 0.f;
}

__global__ void k_negcopy(float* __restrict__ o, const float* __restrict__ s, int n) {
    int i = gtid(); if (i < n) o[i] = -s[i];
}

__global__ void k_addrelu(float* __restrict__ o, const float* __restrict__ a,
                          const float* __restrict__ r, int n) {
    int i = gtid(); if (i < n) o[i] = fmaxf(a[i] + r[i], 0.f);
}

__global__ void k_deg(float* __restrict__ deg, const long long* __restrict__ src) {
    int e = gtid(); if (e < NEDGES) atomicAdd(&deg[src[e]], 1.0f);
}

__global__ void k_dinv(float* __restrict__ deg) {
    int i = gtid();
    if (i < NNODES) { float d = deg[i]; deg[i] = (d > 0.f) ? rsqrtf(d) : 0.f; }
}

__global__ void k_norm(float* __restrict__ norm, const float* __restrict__ dinv,
                       const long long* __restrict__ src, const long long* __restrict__ dst) {
    int e = gtid(); if (e < NEDGES) norm[e] = -dinv[src[e]] * dinv[dst[e]];
}

// out[dst] += scale * norm[e] * x[src], float4 chunks, L2 atomics
__global__ void k_scatter(float* __restrict__ out, const float* __restrict__ x,
                          const long long* __restrict__ src, const long long* __restrict__ dst,
                          const float* __restrict__ norm, int dim4, float scale, int total) {
    int tid = gtid();
    if (tid >= total) return;
    int e = tid / dim4;
    int c = tid - e * dim4;
    long long s = src[e], d = dst[e];
    float w = scale * norm[e];
    float4 v = reinterpret_cast<const float4*>(x)[(long long)s * dim4 + c]; // GLOBAL_LOAD_B128
    float* o = out + ((long long)d * dim4 + c) * 4;
    atomicAdd(o + 0, w * v.x);
    atomicAdd(o + 1, w * v.y);
    atomicAdd(o + 2, w * v.z);
    atomicAdd(o + 3, w * v.w);
}

// One wave computes out[16 x 16] tile = sum_k Tx_k @ W_k (+bias,+res,relu) via
// V_WMMA_F32_16X16X4_F32. Requires din%4==0, dout%16==0, N%16==0 (all true here).
__global__ void k_gemm_wmma(float* __restrict__ out, TxPtrs tx,
                            const float* __restrict__ W, const float* __restrict__ bias,
                            const float* __restrict__ res,
                            int din, int dout, int kterms, int ntilesN, int relu) {
    int wave = gtid() >> 5;
    int lane = threadIdx.x & 31;
    int mt = wave / ntilesN;
    if (mt >= NNODES / 16) return;            // wave-uniform: EXEC stays all-1s
    int nt = wave - mt * ntilesN;
    int half = lane >> 4;                     // which half-wave (K split)
    int lc   = lane & 15;                     // A row / B,C,D column
    int row0 = mt * 16, col0 = nt * 16;
    v8f acc = {0.f, 0.f, 0.f, 0.f, 0.f, 0.f, 0.f, 0.f};
    for (int k = 0; k < kterms; ++k) {
        const float* __restrict__ Xr = tx.p[k] + (size_t)(row0 + lc) * din;
        const float* __restrict__ Wk = W + (size_t)k * din * dout;
        for (int kk = 0; kk < din; kk += 4) {
            int ka = kk + half * 2;
            v2f a, b;
            a.x = Xr[ka];                               // A[m=lc, K=ka]
            a.y = Xr[ka + 1];                           // A[m=lc, K=ka+1]
            b.x = Wk[(size_t)ka * dout + col0 + lc];    // B[K=ka,   n=lc]
            b.y = Wk[(size_t)(ka + 1) * dout + col0 + lc];
            acc = __builtin_amdgcn_wmma_f32_16x16x4_f32(
                false, a, false, b, (short)0, acc, false, false);
        }
    }
    float bv = bias[col0 + lc];
#pragma unroll
    for (int r = 0; r < 8; ++r) {
        int m = row0 + r + half * 8;          // C/D layout: VGPR r -> row r + 8*half
        float v = acc[r] + bv;
        if (res)  v += res[(size_t)m * dout + col0 + lc];
        if (relu) v = fmaxf(v, 0.f);
        out[(size_t)m * dout + col0 + lc] = v;
    }
}

// final 64 -> 2 dense layer (dout not a multiple of 16)
__global__ void k_final(float* __restrict__ out, const float* __restrict__ h,
                        const float* __restrict__ W, const float* __restrict__ bias) {
    int n = gtid();
    if (n >= NNODES) return;
    float a0 = bias[0], a1 = bias[1];
    const float* row = h + (size_t)n * 64;
#pragma unroll 4
    for (int i = 0; i < 64; ++i) {
        float v = row[i];
        a0 += v * W[i * 2 + 0];
        a1 += v * W[i * 2 + 1];
    }
    out[n * 2 + 0] = a0;
    out[n * 2 + 1] = a1;
}

static inline unsigned gdiv(long n, int b) { return (unsigned)((n + b - 1) / b); }

// Full ChebConv: Tx1 = P(in); Tx2 = 2P(Tx1)-in; Tx3 = 2P(Tx2)-Tx1;
// out = act(sum_k Tx_k @ W_k + b). Recurrence folded into scatter init/scale.
static void launch_cheb(hipStream_t st, const float* in, float* out,
                        const float* W, const float* b, int din, int dout, int relu,
                        const long long* src, const long long* dst, const float* norm,
                        float* t1, float* t2, float* t3) {
    int dim4 = din / 4;
    int total = NEDGES * dim4;
    int nel = NNODES * din;
    k_zero   <<<gdiv(nel, 256), 256, 0, st>>>(t1, nel);
    k_scatter<<<gdiv(total, 256), 256, 0, st>>>(t1, in, src, dst, norm, dim4, 1.0f, total);
    k_negcopy<<<gdiv(nel, 256), 256, 0, st>>>(t2, in, nel);
    k_scatter<<<gdiv(total, 256), 256, 0, st>>>(t2, t1, src, dst, norm, dim4, 2.0f, total);
    k_negcopy<<<gdiv(nel, 256), 256, 0, st>>>(t3, t1, nel);
    k_scatter<<<gdiv(total, 256), 256, 0, st>>>(t3, t2, src, dst, norm, dim4, 2.0f, total);
    TxPtrs tx; tx.p[0] = in; tx.p[1] = t1; tx.p[2] = t2; tx.p[3] = t3;
    int ntn = dout / 16;
    long waves = (long)(NNODES / 16) * ntn;
    k_gemm_wmma<<<gdiv(waves, 8), 256, 0, st>>>(out, tx, W, b, nullptr, din, dout, 4, ntn, relu);
}

extern "C" void kernel_launch(void* const* d_in, const int* in_sizes, int n_in,
                              void* d_out, int out_size, void* d_ws, size_t ws_size,
                              hipStream_t stream) {
    (void)in_sizes; (void)n_in; (void)out_size; (void)ws_size;
    const float* x = (const float*)d_in[0];
    const long long* ei = (const long long*)d_in[1];
    const long long* src = ei;
    const long long* dst = ei + NEDGES;

    // pytree leaf order: x, edge_index, 16 blocks x 3 layers x (W,b),
    // 4 res (W,b), 4 reg (W,b)
    const float *blkW[16][3], *blkB[16][3], *resW[4], *resB[4], *regW[4], *regB[4];
    int idx = 2;
    for (int i = 0; i < 16; ++i)
        for (int l = 0; l < 3; ++l) {
            blkW[i][l] = (const float*)d_in[idx++];
            blkB[i][l] = (const float*)d_in[idx++];
        }
    for (int i = 0; i < 4; ++i) { resW[i] = (const float*)d_in[idx++]; resB[i] = (const float*)d_in[idx++]; }
    for (int i = 0; i < 4; ++i) { regW[i] = (const float*)d_in[idx++]; regB[i] = (const float*)d_in[idx++]; }

    // workspace carve-out (floats): everything fits in L2-sized footprint
    float* ws = (float*)d_ws;
    size_t o = 0;
    float* deg  = ws + o; o += 32768;
    float* norm = ws + o; o += NEDGES;
    const size_t NB = (size_t)NNODES * 128;
    float* t1 = ws + o; o += NB;
    float* t2 = ws + o; o += NB;
    float* t3 = ws + o; o += NB;
    float* hA = ws + o; o += NB;
    float* hB = ws + o; o += NB;
    float* hC = ws + o; o += NB;
    float* resbuf = ws + o; o += NB;
    float* big = ws + o; o += (size_t)NNODES * 256;

    // symmetric-normalized Laplacian edge weights
    k_zero<<<gdiv(NNODES, 256), 256, 0, stream>>>(deg, NNODES);
    k_deg <<<gdiv(NEDGES, 256), 256, 0, stream>>>(deg, src);
    k_dinv<<<gdiv(NNODES, 256), 256, 0, stream>>>(deg);
    k_norm<<<gdiv(NEDGES, 256), 256, 0, stream>>>(norm, deg, src, dst);

    static const int BD[16][3] = {
        {16, 16, 32}, {32, 16, 32}, {32, 16, 32}, {32, 32, 64},
        {64, 32, 64}, {64, 32, 64}, {64, 32, 64}, {64, 64, 128},
        {128, 64, 128}, {128, 64, 128}, {128, 64, 128}, {128, 64, 128},
        {128, 64, 128}, {128, 128, 128}, {128, 128, 128}, {128, 128, 128}};

    const float* cur = x;
    float* bufs[3] = {hA, hB, hC};
    for (int blk = 0; blk < 16; ++blk) {
        int din = BD[blk][0], hid = BD[blk][1], dbo = BD[blk][2];
        float* tmp[3]; int nt = 0;
        for (int j = 0; j < 3; ++j) if (bufs[j] != cur) tmp[nt++] = bufs[j];
        int ri = (blk == 0) ? 0 : (blk == 3) ? 1 : (blk == 7) ? 2 : (blk == 13) ? 3 : -1;
        const float* res;
        if (ri >= 0) {
            launch_cheb(stream, cur, resbuf, resW[ri], resB[ri], din, dbo, 0,
                        src, dst, norm, t1, t2, t3);
            res = resbuf;
        } else {
            res = cur;  // preserved: never written during this block
        }
        launch_cheb(stream, cur,    tmp[0], blkW[blk][0], blkB[blk][0], din, hid, 1, src, dst, norm, t1, t2, t3);
        launch_cheb(stream, tmp[0], tmp[1], blkW[blk][1], blkB[blk][1], hid, hid, 1, src, dst, norm, t1, t2, t3);
        launch_cheb(stream, tmp[1], tmp[0], blkW[blk][2], blkB[blk][2], hid, dbo, 1, src, dst, norm, t1, t2, t3);
        int nel = NNODES * dbo;
        k_addrelu<<<gdiv(nel, 256), 256, 0, stream>>>(tmp[1], tmp[0], res, nel);
        cur = tmp[1];
    }

    // regression head: 128->256->128->64->2
    auto dense = [&](const float* in, float* out, const float* W, const float* b,
                     int di, int dq, int relu) {
        TxPtrs tx; tx.p[0] = in; tx.p[1] = in; tx.p[2] = in; tx.p[3] = in;
        int ntn = dq / 16;
        long waves = (long)(NNODES / 16) * ntn;
        k_gemm_wmma<<<gdiv(waves, 8), 256, 0, stream>>>(out, tx, W, b, nullptr, di, dq, 1, ntn, relu);
    };
    dense(cur, big, regW[0], regB[0], 128, 256, 1);
    dense(big, t1,  regW[1], regB[1], 256, 128, 1);
    dense(t1,  t2,  regW[2], regB[2], 128, 64, 1);
    k_final<<<gdiv(NNODES, 256), 256, 0, stream>>>((float*)d_out, t2, regW[3], regB[3]);
}